// NoisyLabelCorrectionVAE_40218073759841
// MI455X (gfx1250) — compile-verified
//
#include <hip/hip_runtime.h>
#include <math.h>

// NoisyLabelCorrectionVAE: per-batch S = V V^T + diag(var), Cholesky, g = L eps,
// Gaussian-copula logistic relaxation.  B=2048, D=256, R=16, f32 end-to-end.
//
// CDNA5 strategy: one workgroup per batch; full 256x256 S in LDS (needs the
// 320KB/WGP LDS of gfx1250); all GEMM-like work (S build + Cholesky trailing
// SYRK updates) via V_WMMA_F32_16X16X4_F32.

typedef __attribute__((ext_vector_type(2))) float v2f;
typedef __attribute__((ext_vector_type(8))) float v8f;

#define DD    256
#define RR    16
#define NB    16
#define TB    (DD / NB)     // 16 block rows/cols
#define LDP   257           // LDS row stride (257 mod 64 == 1 -> conflict-free columns)

__global__ __launch_bounds__(256)
void vae_copula_chol_kernel(const float* __restrict__ y_dist,
                            const float* __restrict__ Vg,
                            const float* __restrict__ log_var,
                            const float* __restrict__ epsg,
                            float* __restrict__ out)
{
    __shared__ float S[DD][LDP];      // lower triangle of S / L  (~263 KB)
    __shared__ float var_s[DD];
    __shared__ float eps_s[DD];

    const int b      = blockIdx.x;
    const int tid    = threadIdx.x;           // 0..255
    const int lane   = tid & 31;              // wave32 lane
    const int wave   = tid >> 5;              // 0..7
    const int nwav   = blockDim.x >> 5;       // 8

    const float* Vb = Vg + (size_t)b * DD * RR;

    // Warm the V panel for the WMMA build (global_prefetch_b8).
    __builtin_prefetch(Vb + (size_t)tid * RR, 0, 3);

    // ---- elementwise preamble: var = clamp(exp(0.5*log_var), 1e-5, 5); stage eps ----
    {
        float lv = log_var[(size_t)b * DD + tid];
        float vv = expf(0.5f * lv);
        var_s[tid] = fminf(fmaxf(vv, 1e-5f), 5.0f);
        eps_s[tid] = epsg[(size_t)b * DD + tid];
    }
    __syncthreads();

    // Fragment lane coordinates (ISA 7.12.2 f32 layouts):
    //  A 16x4:  lanes 0-15 -> M=lane, K-pair {0,1}; lanes 16-31 -> M=lane-16, K-pair {2,3}
    //  B 4x16:  lanes 0-15 -> N=lane, K rows {0,1}; lanes 16-31 -> N=lane-16, K rows {2,3}
    //  C 16x16: VGPR v -> row v (lanes 0-15) / row v+8 (lanes 16-31), col = lane&15
    const int fm = lane & 15;                  // M (A) / N (B) / col (C)
    const int kh = (lane >> 4) << 1;           // 0 or 2: which K pair
    const int r8 = (lane >> 4) << 3;           // 0 or 8: C row offset

    // ---- build lower-triangle S = V V^T with f32 WMMA (diag added separately) ----
    {
        const int nBuild = TB * (TB + 1) / 2;  // 136 blocks
        for (int t = wave; t < nBuild; t += nwav) {
            int i0 = 0;
            while ((i0 + 1) * (i0 + 2) / 2 <= t) ++i0;
            int j0 = t - i0 * (i0 + 1) / 2;

            const float* arow = Vb + (size_t)(i0 * 16 + fm) * RR;
            const float* brow = Vb + (size_t)(j0 * 16 + fm) * RR;

            v8f c = {};
            #pragma unroll
            for (int kc = 0; kc < 4; ++kc) {
                v2f a, bb;
                a.x  = arow[kc * 4 + kh];
                a.y  = arow[kc * 4 + kh + 1];
                bb.x = brow[kc * 4 + kh];
                bb.y = brow[kc * 4 + kh + 1];
                c = __builtin_amdgcn_wmma_f32_16x16x4_f32(
                        false, a, false, bb, (short)0, c, false, false);
            }
            const int rbase = i0 * 16 + r8;
            const int col   = j0 * 16 + fm;
            #pragma unroll
            for (int v = 0; v < 8; ++v)
                S[rbase + v][col] = c[v];          // branch-free store
        }
    }
    __syncthreads();

    // Divergence-free diagonal add: S += diag(var).
    S[tid][tid] += var_s[tid];
    __syncthreads();

    // ---- blocked right-looking Cholesky, NB = 16 ----
    #pragma unroll 1
    for (int kb = 0; kb < TB; ++kb) {
        // Panel factorization: columns kb*16 .. kb*16+15 over rows jc..255.
        // Two barriers per column: (A) previous rank-1 updates visible,
        // (B) everyone captured the column tops before owners overwrite them.
        #pragma unroll
        for (int c = 0; c < NB; ++c) {
            const int jc = kb * NB + c;
            __syncthreads();                               // (A)
            const float dia = S[jc][jc];
            const float inv = 1.0f / sqrtf(dia);

            const int  r   = jc + tid;
            const bool act = (r < DD);
            float lic = 0.0f;
            if (act) lic = S[r][jc] * inv;

            float topv[NB];
            #pragma unroll
            for (int c2 = c + 1; c2 < NB; ++c2)
                topv[c2] = S[kb * NB + c2][jc] * inv;      // scaled column tops

            __syncthreads();                               // (B)
            if (act) {
                S[r][jc] = lic;
                #pragma unroll
                for (int c2 = c + 1; c2 < NB; ++c2)
                    S[r][kb * NB + c2] -= lic * topv[c2];  // rank-1 panel update
            }
        }
        __syncthreads();

        // Trailing SYRK update via WMMA: S(i,j) -= L(i,kb) * L(j,kb)^T,
        // for kb < j <= i < 16.  Minus folded into the A fragment (f32 WMMA
        // NEG only supports C-negate per ISA 7.12).
        const int mrows = TB - 1 - kb;
        const int nblk  = mrows * (mrows + 1) / 2;
        for (int t = wave; t < nblk; t += nwav) {          // wave-uniform: EXEC all 1s
            int i0 = 0;
            while ((i0 + 1) * (i0 + 2) / 2 <= t) ++i0;
            int j0 = t - i0 * (i0 + 1) / 2;
            const int bi = kb + 1 + i0;
            const int bj = kb + 1 + j0;

            const int rbase = bi * 16 + r8;
            const int colc  = bj * 16 + fm;

            v8f cc;
            #pragma unroll
            for (int v = 0; v < 8; ++v) cc[v] = S[rbase + v][colc];

            #pragma unroll
            for (int kc = 0; kc < 4; ++kc) {
                const int kcol = kb * 16 + kc * 4 + kh;
                v2f a, bb;
                a.x  = -S[bi * 16 + fm][kcol];
                a.y  = -S[bi * 16 + fm][kcol + 1];
                bb.x =  S[bj * 16 + fm][kcol];
                bb.y =  S[bj * 16 + fm][kcol + 1];
                cc = __builtin_amdgcn_wmma_f32_16x16x4_f32(
                        false, a, false, bb, (short)0, cc, false, false);
            }
            #pragma unroll
            for (int v = 0; v < 8; ++v) S[rbase + v][colc] = cc[v];
        }
    }
    __syncthreads();

    // ---- g = L * eps (row-parallel lower-triangular matvec) + copula tail ----
    {
        const int r = tid;
        float acc = 0.0f;
        for (int c = 0; c <= r; ++c)
            acc = fmaf(S[r][c], eps_s[c], acc);

        const float x = acc / var_s[r];
        float u = 0.5f * erfcf(-x * 0.70710678118654752440f);   // ndtr(x)
        u = fminf(fmaxf(u, 1e-6f), 0.999f);

        float yd = y_dist[(size_t)b * DD + r];
        yd = fminf(fmaxf(yd, 1e-6f), 0.999f);

        const float z = logf(yd) - logf(1.0f - yd) + logf(u) - logf(1.0f - u);
        const float y = 1.0f / (1.0f + expf(-2.0f * z));        // sigmoid(z / 0.5)

        out[(size_t)b * DD + r] = y;
    }
}

extern "C" void kernel_launch(void* const* d_in, const int* in_sizes, int n_in,
                              void* d_out, int out_size, void* d_ws, size_t ws_size,
                              hipStream_t stream)
{
    const float* y_dist  = (const float*)d_in[0];
    const float* V       = (const float*)d_in[1];
    const float* log_var = (const float*)d_in[2];
    const float* eps     = (const float*)d_in[3];
    float*       out     = (float*)d_out;

    const int B = in_sizes[0] / DD;   // 2048
    hipLaunchKernelGGL(vae_copula_chol_kernel, dim3(B), dim3(DD), 0, stream,
                       y_dist, V, log_var, eps, out);
}